// LocalPODLoss_71579924955582
// MI455X (gfx1250) — compile-verified
//
#include <hip/hip_runtime.h>
#include <math.h>

typedef __attribute__((ext_vector_type(2))) float v2f;
typedef __attribute__((ext_vector_type(8))) float v8f;

#define STRIDE   36              // padded row stride (floats): 144B rows keep 16B alignment
#define TILE_LDS (32 * STRIDE)   // one 32x32 tile per wave
#define WAVES    8               // 256-thread block = 8 waves (wave32)

// One wave processes one (b,c) 32x32 tile:
//   D = new - old  ->  LDS
//   R = Band(16x32) x D(32x32)   via v_wmma_f32_16x16x4_f32 (2 output tiles x 8 K-steps)
//   C = D(32x32)   x Band^T(32x16)                          (2 output tiles x 8 K-steps)
//   accumulate cnt-weighted squares, reduce to one float per block.
__global__ __launch_bounds__(256) void pod_main_kernel(const float* __restrict__ nf,
                                                       const float* __restrict__ of,
                                                       float* __restrict__ partial,
                                                       int ntiles) {
    __shared__ __align__(16) float smem[WAVES * TILE_LDS];
    __shared__ float wred[WAVES];

    const int wave = threadIdx.x >> 5;
    const int lane = threadIdx.x & 31;
    const int half = lane >> 4;       // 0: lanes 0-15, 1: lanes 16-31
    const int lm   = lane & 15;
    const int tile = blockIdx.x * WAVES + wave;

    float* T = &smem[wave * TILE_LDS];
    float wsum = 0.0f;

    if (tile < ntiles) {              // wave-uniform guard -> EXEC all ones inside
        const float* np = nf + (size_t)tile * 1024;
        const float* op = of + (size_t)tile * 1024;

        // Stream 4KB tile of each input, diff, stage into LDS (aligned b128 stores).
#pragma unroll
        for (int it = 0; it < 8; ++it) {
            int e = it * 128 + lane * 4;
            float4 a = *(const float4*)(np + e);
            float4 b = *(const float4*)(op + e);
            float4 d;
            d.x = a.x - b.x; d.y = a.y - b.y; d.z = a.z - b.z; d.w = a.w - b.w;
            int r = e >> 5, c = e & 31;
            *(float4*)(&T[r * STRIDE + c]) = d;
        }
        // Same-wave LDS RAW: wait on DS counter only (no cross-wave sharing).
        asm volatile("s_wait_dscnt 0" ::: "memory");

        v8f r0 = {0.f,0.f,0.f,0.f,0.f,0.f,0.f,0.f};
        v8f r1 = {0.f,0.f,0.f,0.f,0.f,0.f,0.f,0.f};
        v8f c0 = {0.f,0.f,0.f,0.f,0.f,0.f,0.f,0.f};
        v8f c1 = {0.f,0.f,0.f,0.f,0.f,0.f,0.f,0.f};

#pragma unroll
        for (int k = 0; k < 8; ++k) {
            const int x0 = 4 * k + 2 * half;   // K index this lane-half covers (v=0)
            // Banded-ones operand, identical register layout as A (16xK) and B (Kx16):
            // band[v] = 1 iff 0 <= (x0+v) - lm <= 15
            v2f band;
            band.x = ((unsigned)(x0     - lm) <= 15u) ? 1.0f : 0.0f;
            band.y = ((unsigned)(x0 + 1 - lm) <= 15u) ? 1.0f : 0.0f;

            // B operand for R = Band x T : rows x0,x0+1 ; cols lm / 16+lm
            v2f b0, b1;
            b0.x = T[ x0      * STRIDE + lm];
            b0.y = T[(x0 + 1) * STRIDE + lm];
            b1.x = T[ x0      * STRIDE + 16 + lm];
            b1.y = T[(x0 + 1) * STRIDE + 16 + lm];

            // A operand for C = T x Band^T : row lm / 16+lm ; cols x0,x0+1 (contiguous b64)
            v2f a0 = *(const v2f*)(&T[ lm       * STRIDE + x0]);
            v2f a1 = *(const v2f*)(&T[(16 + lm) * STRIDE + x0]);

            r0 = __builtin_amdgcn_wmma_f32_16x16x4_f32(false, band, false, b0, (short)0, r0, false, false);
            r1 = __builtin_amdgcn_wmma_f32_16x16x4_f32(false, band, false, b1, (short)0, r1, false, false);
            c0 = __builtin_amdgcn_wmma_f32_16x16x4_f32(false, a0, false, band, (short)0, c0, false, false);
            c1 = __builtin_amdgcn_wmma_f32_16x16x4_f32(false, a1, false, band, (short)0, c1, false, false);
        }

        // cnt-weighted square accumulation using C/D layout:
        //   lane -> N = lm (+16 for tile1), VGPR r -> M = r + 8*half.
        // R tiles: weight depends on column y -> per-lane scalar.
        const float wr0 = (float)(lm + 1);     // y = lm,        cnt = y+1
        const float wr1 = (float)(15 - lm);    // y = 16+lm,     cnt = 31-y
        float s0 = 0.0f, s1 = 0.0f;
#pragma unroll
        for (int r = 0; r < 8; ++r) {
            s0 += r0[r] * r0[r];
            s1 += r1[r] * r1[r];
            const float wc0 = (float)(r + 8 * half + 1);   // x = r+8*half,    cnt = x+1
            const float wc1 = (float)(15 - r - 8 * half);  // x = 16+r+8*half, cnt = 31-x
            wsum += wc0 * c0[r] * c0[r] + wc1 * c1[r] * c1[r];
        }
        wsum += wr0 * s0 + wr1 * s1;
    }

    // wave32 butterfly reduce
#pragma unroll
    for (int off = 16; off > 0; off >>= 1)
        wsum += __shfl_xor(wsum, off, 32);
    if (lane == 0) wred[wave] = wsum;
    __syncthreads();
    if (threadIdx.x == 0) {
        float s = 0.0f;
#pragma unroll
        for (int i = 0; i < WAVES; ++i) s += wred[i];
        partial[blockIdx.x] = s;
    }
}

// Deterministic final reduction + sqrt/scale.
__global__ __launch_bounds__(256) void pod_final_kernel(const float* __restrict__ partial,
                                                        int n, float* __restrict__ out) {
    __shared__ float wred[WAVES];
    float s = 0.0f;
    for (int i = threadIdx.x; i < n; i += 256) s += partial[i];
#pragma unroll
    for (int off = 16; off > 0; off >>= 1)
        s += __shfl_xor(s, off, 32);
    const int wave = threadIdx.x >> 5;
    const int lane = threadIdx.x & 31;
    if (lane == 0) wred[wave] = s;
    __syncthreads();
    if (threadIdx.x == 0) {
        float t = 0.0f;
#pragma unroll
        for (int i = 0; i < WAVES; ++i) t += wred[i];
        // ss = t/256 ; loss = 1e-6 + sqrt(ss) ; out = ALPHA*loss/2
        out[0] = 0.5f * (1e-6f + sqrtf(t) * (1.0f / 16.0f));
    }
}

extern "C" void kernel_launch(void* const* d_in, const int* in_sizes, int n_in,
                              void* d_out, int out_size, void* d_ws, size_t ws_size,
                              hipStream_t stream) {
    const float* nf = (const float*)d_in[0];
    const float* of = (const float*)d_in[1];
    float* partial = (float*)d_ws;
    float* out = (float*)d_out;

    const int total  = in_sizes[0];            // B*C*W*H = 16*512*32*32
    const int ntiles = total >> 10;            // 32x32 tiles = 8192
    const int nblocks = (ntiles + WAVES - 1) / WAVES;

    pod_main_kernel<<<nblocks, 256, 0, stream>>>(nf, of, partial, ntiles);
    pod_final_kernel<<<1, 256, 0, stream>>>(partial, nblocks, out);
}